// ScalarBorn_86973087744667
// MI455X (gfx1250) — compile-verified
//
#include <hip/hip_runtime.h>
#include <hip/hip_bf16.h>
#include <stdint.h>

typedef __attribute__((ext_vector_type(2))) float v2f;
typedef __attribute__((ext_vector_type(8))) float v8f;

// ---------------- problem constants ----------------
constexpr int   NYv   = 200, NXv = 200;
constexpr int   PMLn  = 20, FDP = 2, PADp = 22;
constexpr int   NPAD  = 244;                 // padded grid dim
constexpr int   NPLANE= NPAD*NPAD;           // 59536
constexpr float DTc   = 0.0005f;
constexpr float Hc    = 4.0f;                // DX == DY
constexpr int   NTt   = 120;
constexpr int   NSHOT = 2, NRECn = 100;
constexpr int   TILEc = 32, TPDc = 8;
constexpr int   NBLKc = NSHOT*TPDc*TPDc;     // 128 persistent blocks

// ---------------- workspace layout (float units) ----------------
constexpr int OFF_V2    = 0;                    // v^2 dt^2          (NPLANE)
constexpr int OFF_BORN  = NPLANE;               // 2 v dt^2 scatter  (NPLANE)
constexpr int OFF_AY    = 2*NPLANE;             // PML profiles      (NPAD each)
constexpr int OFF_BY    = OFF_AY + NPAD;
constexpr int OFF_AX    = OFF_BY + NPAD;
constexpr int OFF_BX    = OFF_AX + NPAD;
constexpr int OFF_CNT   = OFF_BX + NPAD + 8;    // grid-barrier counter
constexpr int OFF_STATE = OFF_CNT + 8;
constexpr int FSTR      = NSHOT*NPLANE;         // per-field stride
// fields: 0 wf,1 psiy,2 psix,3 zy,4 zx (bg) ; +5 for scattered
constexpr int STATE_SZ  = 10*FSTR;              // one buffer; x2 (double buffered)
// total ws: OFF_STATE + 2*STATE_SZ floats  ~= 10 MB

static __device__ inline v8f wmma4(v2f a, v2f b, v8f c) {
  // D = A(16x4) x B(4x16) + C   (f32, wave32)
  return __builtin_amdgcn_wmma_f32_16x16x4_f32(false, a, false, b, (short)0, c, false, false);
}

// ---------------- init: max(v) reduction + CPML profiles ----------------
__global__ void born_prep(const float* __restrict__ v, float* __restrict__ ws) {
  __shared__ float red[1024];
  int tid = threadIdx.x;
  float m = 0.f;
  for (int i = tid; i < NYv*NXv; i += 1024) m = fmaxf(m, v[i]);
  red[tid] = m; __syncthreads();
  for (int s = 512; s > 0; s >>= 1) {
    if (tid < s) red[tid] = fmaxf(red[tid], red[tid+s]);
    __syncthreads();
  }
  float maxv = red[0];
  if (tid < NPAD) {
    float x  = (float)tid;
    float lo = (float)(FDP + PMLn);
    float hi = (float)(NPAD - 1 - FDP - PMLn);
    float dA = fminf(fmaxf((lo - x)/(float)PMLn, 0.f), 1.f);
    float dB = fminf(fmaxf((x - hi)/(float)PMLn, 0.f), 1.f);
    float d  = fmaxf(dA, dB);
    float sigma = 3.f*maxv*6.90775527898f/(2.f*(float)PMLn*Hc)*d*d;
    float alpha = 2.f*3.14159265359f*25.f*(1.f - d);
    float a = expf(-(sigma+alpha)*DTc);
    float b = sigma/(sigma+alpha+1e-9f)*(a-1.f);
    ws[OFF_AY+tid]=a; ws[OFF_BY+tid]=b; ws[OFF_AX+tid]=a; ws[OFF_BX+tid]=b;
  }
  if (tid == 0) *((unsigned*)(ws+OFF_CNT)) = 0u;
}

// ---------------- init: padded coeff planes + zero state ----------------
__global__ void born_fill(const float* __restrict__ v, const float* __restrict__ sc,
                          float* __restrict__ ws) {
  long stride = (long)gridDim.x*blockDim.x;
  long tid0   = (long)blockIdx.x*blockDim.x + threadIdx.x;
  for (long i = tid0; i < 2L*STATE_SZ; i += stride) ws[OFF_STATE + i] = 0.f;
  for (long i = tid0; i < NPLANE; i += stride) {
    int y = (int)(i / NPAD), x = (int)(i % NPAD);
    int uy = min(max(y-PADp,0),NYv-1), ux = min(max(x-PADp,0),NXv-1);
    float vv = v[uy*NXv+ux];
    ws[OFF_V2+i] = vv*vv*DTc*DTc;
    bool in = (y>=PADp && y<PADp+NYv && x>=PADp && x<PADp+NXv);
    float sp = in ? sc[(y-PADp)*NXv + (x-PADp)] : 0.f;
    ws[OFF_BORN+i] = 2.f*vv*DTc*DTc*sp;
  }
}

// ---------------- persistent propagation kernel ----------------
extern "C" __global__ void __launch_bounds__(256, 1)
born_persist(const float* __restrict__ amp, const int* __restrict__ sloc,
             const int* __restrict__ rloc, float* __restrict__ ws,
             float* __restrict__ out) {
  __shared__ float wB[40*41];     // background wavefield tile (halo 4)
  __shared__ float wS[40*41];     // scattered  wavefield tile
  __shared__ float pyB[36*37], pxB[36*37];  // psi-new, bg (halo 2)
  __shared__ float pyS[36*37], pxS[36*37];  // psi-new, sc
  __shared__ float lapB[32*33];   // background laplacian for Born coupling

  const int tid  = threadIdx.x;
  const int lane = tid & 31;
  const int wv   = tid >> 5;
  const int b    = blockIdx.x;
  const int s    = b / (TPDc*TPDc);
  const int t6   = b % (TPDc*TPDc);
  const int gy0  = (t6 / TPDc) * TILEc;
  const int gx0  = (t6 % TPDc) * TILEc;

  const float* v2   = ws + OFF_V2;
  const float* born = ws + OFF_BORN;
  const float* pay  = ws + OFF_AY;  const float* pby = ws + OFF_BY;
  const float* pax  = ws + OFF_AX;  const float* pbx = ws + OFF_BX;
  unsigned* cnt = (unsigned*)(ws + OFF_CNT);
  float* stA = ws + OFF_STATE;
  float* stB = stA + STATE_SZ;

  const int sly = sloc[s*2+0] + PADp, slx = sloc[s*2+1] + PADp;

  // --- constant banded-coefficient WMMA fragments (per-lane, live in VGPRs) ---
  const int half = lane >> 4, lidx = lane & 15;
  const float d1c[5] = { 1.f/12.f, -2.f/3.f, 0.f, 2.f/3.f, -1.f/12.f };
  const float d2c[5] = { -1.f/12.f, 4.f/3.f, -2.5f, 4.f/3.f, -1.f/12.f };
  v2f cb2[5], cb1[5];
  #pragma unroll
  for (int k = 0; k < 5; ++k) {
    #pragma unroll
    for (int p = 0; p < 2; ++p) {
      int j  = 4*k + 2*half + p;
      int dm = j - lidx;
      float c2 = (dm >= 0 && dm <= 4) ? d2c[dm]*(1.f/(Hc*Hc)) : 0.f;
      float c1 = (dm >= 0 && dm <= 4) ? d1c[dm]*(1.f/Hc)      : 0.f;
      cb2[k][p] = c2; cb1[k][p] = c1;
    }
  }

  // per-wave phase-C assignment: 8 waves = 2 fields x 4 subtiles
  const int f    = wv >> 2;              // 0 = background, 1 = scattered
  const int sub  = wv & 3;
  const int sm0  = (sub >> 1) * 16;
  const int sn0  = (sub & 1) * 16;
  float* wlds = f ? wS  : wB;
  float* pyl  = f ? pyS : pyB;
  float* pxl  = f ? pxS : pxB;

  for (int t = 0; t < NTt; ++t) {
    float* cur = (t & 1) ? stB : stA;
    float* nxt = (t & 1) ? stA : stB;

    // receivers for step t-1 from read-only cur scattered field
    if (b == 0 && t > 0 && tid < NSHOT*NRECn) {
      int rs = tid / NRECn;
      int ry = rloc[tid*2+0] + PADp, rx = rloc[tid*2+1] + PADp;
      out[tid*NTt + (t-1)] = cur[5*FSTR + rs*NPLANE + ry*NPAD + rx];
    }

    // ---- Phase A: async-stage both 40x40 wavefield tiles into LDS ----
    {
      const float* wg0 = cur + 0*FSTR + s*NPLANE;
      const float* wg1 = cur + 5*FSTR + s*NPLANE;
      unsigned lb0 = (unsigned)(uintptr_t)wB;
      unsigned lb1 = (unsigned)(uintptr_t)wS;
      for (int i = tid; i < 40*40; i += 256) {
        int r = i / 40, c = i % 40;
        int gy = gy0 - 4 + r, gx = gx0 - 4 + c;
        unsigned lo = (unsigned)((r*41 + c)*4);
        if ((unsigned)gy < (unsigned)NPAD && (unsigned)gx < (unsigned)NPAD) {
          const float* ga0 = wg0 + gy*NPAD + gx;
          const float* ga1 = wg1 + gy*NPAD + gx;
          unsigned a0 = lb0 + lo, a1 = lb1 + lo;
          asm volatile("global_load_async_to_lds_b32 %0, %1, off"
                       :: "v"(a0), "v"(ga0) : "memory");
          asm volatile("global_load_async_to_lds_b32 %0, %1, off"
                       :: "v"(a1), "v"(ga1) : "memory");
        } else {
          wB[r*41+c] = 0.f;
          wS[r*41+c] = 0.f;
        }
      }
      asm volatile("s_wait_asynccnt 0" ::: "memory");
    }
    __syncthreads();

    // ---- Phase B: psi-new on 36x36 halo region (both fields, both dirs) ----
    {
      const float* cpyB = cur + 1*FSTR + s*NPLANE;
      const float* cpxB = cur + 2*FSTR + s*NPLANE;
      const float* cpyS = cur + 6*FSTR + s*NPLANE;
      const float* cpxS = cur + 7*FSTR + s*NPLANE;
      float* npyB = nxt + 1*FSTR + s*NPLANE;
      float* npxB = nxt + 2*FSTR + s*NPLANE;
      float* npyS = nxt + 6*FSTR + s*NPLANE;
      float* npxS = nxt + 7*FSTR + s*NPLANE;
      const float k1 = 1.f/(12.f*Hc), k2 = 2.f/(3.f*Hc);
      for (int i = tid; i < 36*36; i += 256) {
        int r = i / 36, c = i % 36;
        int gy = gy0 - 2 + r, gx = gx0 - 2 + c;
        bool in = ((unsigned)gy < (unsigned)NPAD) && ((unsigned)gx < (unsigned)NPAD);
        int gyc = min(max(gy,0),NPAD-1), gxc = min(max(gx,0),NPAD-1);
        int gi  = gyc*NPAD + gxc;
        float ayv = pay[gyc], byv = pby[gyc];
        float axv = pax[gxc], bxv = pbx[gxc];
        int wb = (r+2)*41 + (c+2);
        // background
        float d1y = k1*(wB[wb-82]-wB[wb+82]) + k2*(wB[wb+41]-wB[wb-41]);
        float d1x = k1*(wB[wb-2]-wB[wb+2])   + k2*(wB[wb+1]-wB[wb-1]);
        float pyn = in ? (ayv*cpyB[gi] + byv*d1y) : 0.f;
        float pxn = in ? (axv*cpxB[gi] + bxv*d1x) : 0.f;
        pyB[r*37+c] = pyn;  pxB[r*37+c] = pxn;
        // scattered
        float d1ys = k1*(wS[wb-82]-wS[wb+82]) + k2*(wS[wb+41]-wS[wb-41]);
        float d1xs = k1*(wS[wb-2]-wS[wb+2])   + k2*(wS[wb+1]-wS[wb-1]);
        float pyns = in ? (ayv*cpyS[gi] + byv*d1ys) : 0.f;
        float pxns = in ? (axv*cpxS[gi] + bxv*d1xs) : 0.f;
        pyS[r*37+c] = pyns; pxS[r*37+c] = pxns;
        bool inner = (r>=2 && r<34 && c>=2 && c<34);
        if (in && inner) {
          npyB[gi]=pyn;  npxB[gi]=pxn;
          npyS[gi]=pyns; npxS[gi]=pxns;
        }
      }
    }
    __syncthreads();

    // ---- Phase C: per-wave WMMA stencil + leapfrog/Born epilogue ----
    {
      const float* czy = cur + (f*5+3)*FSTR + s*NPLANE;
      const float* czx = cur + (f*5+4)*FSTR + s*NPLANE;
      float* nzy = nxt + (f*5+3)*FSTR + s*NPLANE;
      float* nzx = nxt + (f*5+4)*FSTR + s*NPLANE;
      float* nwf = nxt + (f*5+0)*FSTR + s*NPLANE;

      // ty = d2y(w) + d1y(psiy_new): 10 chained V_WMMA_F32_16X16X4_F32
      v8f ty = {};
      #pragma unroll
      for (int k = 0; k < 5; ++k) {
        int rr = sm0 + 2 + 4*k + 2*half;
        v2f bw; bw[0] = wlds[rr*41 + (sn0+4+lidx)];
                bw[1] = wlds[(rr+1)*41 + (sn0+4+lidx)];
        ty = wmma4(cb2[k], bw, ty);
      }
      #pragma unroll
      for (int k = 0; k < 5; ++k) {
        int rr = sm0 + 4*k + 2*half;
        v2f bp; bp[0] = pyl[rr*37 + (sn0+2+lidx)];
                bp[1] = pyl[(rr+1)*37 + (sn0+2+lidx)];
        ty = wmma4(cb1[k], bp, ty);
      }
      // tx = d2x(w) + d1x(psix_new)
      v8f tx = {};
      #pragma unroll
      for (int k = 0; k < 5; ++k) {
        int cc = sn0 + 2 + 4*k + 2*half;
        v2f aw; aw[0] = wlds[(sm0+4+lidx)*41 + cc];
                aw[1] = wlds[(sm0+4+lidx)*41 + cc + 1];
        tx = wmma4(aw, cb2[k], tx);
      }
      #pragma unroll
      for (int k = 0; k < 5; ++k) {
        int cc = sn0 + 4*k + 2*half;
        v2f ap; ap[0] = pxl[(sm0+2+lidx)*37 + cc];
                ap[1] = pxl[(sm0+2+lidx)*37 + cc + 1];
        tx = wmma4(ap, cb1[k], tx);
      }

      float part[8];
      bool  ok[8];
      int   gidx[8];
      #pragma unroll
      for (int r = 0; r < 8; ++r) {
        int m = r + 8*half, n = lidx;
        int gy = gy0 + sm0 + m, gx = gx0 + sn0 + n;
        bool in = (gy < NPAD) && (gx < NPAD);
        ok[r] = in; part[r] = 0.f; gidx[r] = 0;
        if (in) {
          int gi = gy*NPAD + gx;
          gidx[r] = gi;
          float tyv = ty[r], txv = tx[r];
          float zyn = pay[gy]*czy[gi] + pby[gy]*tyv;
          float zxn = pax[gx]*czx[gi] + pbx[gx]*txv;
          float lap = tyv + zyn + txv + zxn;
          float wfc = wlds[(sm0+4+m)*41 + (sn0+4+n)];
          float wfp = nwf[gi];                 // old buffer value == wf(t-1)
          float wfn = v2[gi]*lap + 2.f*wfc - wfp;
          nzy[gi] = zyn; nzx[gi] = zxn;
          if (f == 0) {
            if (gy == sly && gx == slx) wfn += v2[gi]*amp[s*NTt + t];
            lapB[(sm0+m)*33 + (sn0+n)] = lap;  // publish for Born coupling
            nwf[gi] = wfn;
          } else {
            part[r] = wfn;
          }
        }
      }
      __syncthreads();
      if (f == 1) {
        #pragma unroll
        for (int r = 0; r < 8; ++r) {
          if (ok[r]) {
            int m = r + 8*half, n = lidx;
            nwf[gidx[r]] = part[r] + born[gidx[r]]*lapB[(sm0+m)*33 + (sn0+n)];
          }
        }
      }
    }

    // ---- device-wide barrier (monotonic counter, blocks are co-resident) ----
    __syncthreads();
    if (tid == 0) {
      __threadfence();
      __hip_atomic_fetch_add(cnt, 1u, __ATOMIC_RELEASE, __HIP_MEMORY_SCOPE_AGENT);
      unsigned tgt = (unsigned)(t+1) * (unsigned)NBLKc;
      while (__hip_atomic_load(cnt, __ATOMIC_ACQUIRE, __HIP_MEMORY_SCOPE_AGENT) < tgt) {
        __builtin_amdgcn_s_sleep(2);
      }
      __threadfence();
    }
    __syncthreads();
  }

  // final receiver sample (step NT-1)
  if (b == 0 && tid < NSHOT*NRECn) {
    float* fin = ((NTt-1) & 1) ? stA : stB;   // == nxt of last step
    int rs = tid / NRECn;
    int ry = rloc[tid*2+0] + PADp, rx = rloc[tid*2+1] + PADp;
    out[tid*NTt + (NTt-1)] = fin[5*FSTR + rs*NPLANE + ry*NPAD + rx];
  }
}

// ---------------- host entry ----------------
extern "C" void kernel_launch(void* const* d_in, const int* in_sizes, int n_in,
                              void* d_out, int out_size, void* d_ws, size_t ws_size,
                              hipStream_t stream) {
  (void)in_sizes; (void)n_in; (void)out_size; (void)ws_size;
  const float* v    = (const float*)d_in[0];
  const float* sc   = (const float*)d_in[1];
  const float* amp  = (const float*)d_in[2];
  const int*   sloc = (const int*)d_in[3];
  const int*   rloc = (const int*)d_in[4];
  float* ws  = (float*)d_ws;   // needs OFF_STATE + 2*STATE_SZ floats (~10 MB)
  float* out = (float*)d_out;

  born_prep<<<1, 1024, 0, stream>>>(v, ws);
  born_fill<<<1024, 256, 0, stream>>>(v, sc, ws);
  born_persist<<<NBLKc, 256, 0, stream>>>(amp, sloc, rloc, ws, out);
}